// HistogramLoss_43207370998095
// MI455X (gfx1250) — compile-verified
//
#include <hip/hip_runtime.h>
#include <hip/hip_bf16.h>
#include <stdint.h>

#define HDIM 2048
#define HH (HDIM * HDIM)
#define BINS 256
#define NHIST 6

// workspace layout (bytes)
#define HIST_OFF   0
#define HIST_BYTES (NHIST * BINS * 4)          // 6144
#define TAB_OFF    (HIST_OFF + HIST_BYTES)     // 6144
#define TAB_BYTES  (3 * BINS * 4)              // 3072
#define NPART      2048
#define PART_OFF   (TAB_OFF + TAB_BYTES)       // 9216
#define PART_BYTES (NPART * 4)                 // 8192
#define FLAG_OFF   (PART_OFF + PART_BYTES)     // 17408
#define FLAG_WORDS (HH / 32)                   // 131072
#define FLAG_BYTES (FLAG_WORDS * 4)            // 524288
#define WS_TOTAL   (FLAG_OFF + FLAG_BYTES)     // ~541 KB

typedef __attribute__((ext_vector_type(2))) float v2f;
typedef __attribute__((ext_vector_type(8))) float v8f;

__device__ __forceinline__ float img255(float x) {
    // matches reference op order: ((x + 1) * 0.5) * 255
    return ((x + 1.0f) * 0.5f) * 255.0f;
}

__device__ __forceinline__ int bin256(float v) {
    int b = (int)floorf(v);
    b = b < 0 ? 0 : b;
    return b > 255 ? 255 : b;
}

// ---------------------------------------------------------------------------
// Kernel 1: gathered histograms (LDS u32 atomics, deterministic) + touch flags
// ---------------------------------------------------------------------------
__global__ void hist_flags_kernel(const int* __restrict__ i0, const int* __restrict__ i1,
                                  const int* __restrict__ i2, const int* __restrict__ i3,
                                  const float* __restrict__ ref, const float* __restrict__ tgt,
                                  const float* __restrict__ msrc, const float* __restrict__ mtar,
                                  unsigned int* __restrict__ g_hist,
                                  unsigned int* __restrict__ flag_words, int n) {
    __shared__ unsigned int sh[NHIST * BINS];
    for (int i = threadIdx.x; i < NHIST * BINS; i += blockDim.x) sh[i] = 0u;
    __syncthreads();

    for (int k = blockIdx.x * blockDim.x + threadIdx.x; k < n;
         k += gridDim.x * blockDim.x) {
        int a0 = i0[k], a1 = i1[k], a2 = i2[k], a3 = i3[k];
        int p01 = (a0 << 11) + a1;       // HDIM == 2048
        int p23 = (a2 << 11) + a3;
        float ms = msrc[p01];
        float mt = mtar[p23];
#pragma unroll
        for (int c = 0; c < 3; ++c) {
            float dv = img255(ref[c * HH + p01]) * ms;   // dst_align (masked ref)
            atomicAdd(&sh[c * BINS + bin256(dv)], 1u);
            float rv = img255(tgt[c * HH + p23]) * mt;   // ref_align (masked target)
            atomicAdd(&sh[(3 + c) * BINS + bin256(rv)], 1u);
        }
        atomicOr(&flag_words[p01 >> 5], 1u << (p01 & 31));
    }
    __syncthreads();
    for (int i = threadIdx.x; i < NHIST * BINS; i += blockDim.x) {
        unsigned int v = sh[i];
        if (v) atomicAdd(&g_hist[i], v);
    }
}

// ---------------------------------------------------------------------------
// Kernel 2: 256-bin inclusive prefix sums via chained V_WMMA_F32_16X16X4_F32,
// then CDFs and the 3 transfer tables.  Single block of 256 threads.
//
// Scan-by-matmul: hist as 16x16 Hm.  P = Hm x U (U upper-tri ones, K=16 as
// 4 chained K=4 WMMAs).  Then D = Lstrict x broadcast(rowsum) + P (second
// chain, P reused as the C accumulator) = full 256-element inclusive scan.
// f32 WMMA keeps integer counts (< 2^24) exact.
// ---------------------------------------------------------------------------
__global__ void cdf_table_kernel(const unsigned int* __restrict__ g_hist,
                                 float* __restrict__ g_tables) {
    __shared__ float s_hist[NHIST][BINS];
    __shared__ float s_cdf[NHIST][BINS];

    int t = threadIdx.x;
    for (int i = t; i < NHIST * BINS; i += blockDim.x)
        s_hist[i >> 8][i & 255] = (float)g_hist[i];
    __syncthreads();

    if (t < 32) {                       // wave 0 only; EXEC all-ones in-wave
        const int l  = t;
        const int m  = l & 15;          // A-matrix row
        const int nn = l & 15;          // B/D column
        const bool hi = l >= 16;

        for (int h = 0; h < NHIST; ++h) {
            v8f c = {};
            // ---- chain 1: P = Hm x U (row-wise inclusive prefix) ----
#pragma unroll
            for (int q = 0; q < 4; ++q) {
                const int kx = 4 * q + (hi ? 2 : 0);
                const int ky = 4 * q + (hi ? 3 : 1);
                v2f a, b;
                a.x = s_hist[h][16 * m + kx];
                a.y = s_hist[h][16 * m + ky];
                b.x = (kx <= nn) ? 1.0f : 0.0f;   // U upper-triangular ones
                b.y = (ky <= nn) ? 1.0f : 0.0f;
                c = __builtin_amdgcn_wmma_f32_16x16x4_f32(
                        false, a, false, b, (short)0, c, false, false);
            }
            // ---- pull rowsum[r] = P[r][15] out of the D layout ----
            union { v8f v; float f[8]; } cu; cu.v = c;
            float rs[16];
#pragma unroll
            for (int r = 0; r < 8; ++r) {
                rs[r]     = __shfl(cu.f[r], 15, 32);   // rows 0..7  -> lane 15
                rs[r + 8] = __shfl(cu.f[r], 31, 32);   // rows 8..15 -> lane 31
            }
            // ---- chain 2: D = Lstrict x broadcast(rowsum) + P ----
#pragma unroll
            for (int q = 0; q < 4; ++q) {
                const int kx = 4 * q + (hi ? 2 : 0);
                const int ky = 4 * q + (hi ? 3 : 1);
                v2f a, b;
                a.x = (kx < m) ? 1.0f : 0.0f;          // strict lower-tri ones
                a.y = (ky < m) ? 1.0f : 0.0f;
                b.x = hi ? rs[4 * q + 2] : rs[4 * q + 0];
                b.y = hi ? rs[4 * q + 3] : rs[4 * q + 1];
                c = __builtin_amdgcn_wmma_f32_16x16x4_f32(
                        false, a, false, b, (short)0, c, false, false);
            }
            cu.v = c;
            float total = __shfl(cu.f[7], 31, 32);     // cumsum[255] == N
            float inv = 1.0f / total;
#pragma unroll
            for (int v = 0; v < 8; ++v) {
                int row = v + (hi ? 8 : 0);
                s_cdf[h][16 * row + nn] = cu.f[v] * inv;
            }
        }
    }
    __syncthreads();

    // transfer tables: trans_table(cdf_dst=s_cdf[c], cdf_ref=s_cdf[3+c])
    for (int c = 0; c < 3; ++c) {
        float val;
        if (t == 0)        val = 0.0f;
        else if (t == 255) val = 255.0f;
        else {
            float iv = s_cdf[c][t];              // i_vals[t-1] = cdf_dst[t]
            const float* adj = s_cdf[3 + c];
            int res = t;                          // fallback: identity
            for (int j = 0; j < 255; ++j) {
                if (adj[j] <= iv && iv <= adj[j + 1]) { res = j + 1; break; }
            }
            val = (float)res;
        }
        g_tables[c * BINS + t] = val;
    }
}

// ---------------------------------------------------------------------------
// Kernel 3: fused MSE pass.  matched == masked-ref except at flagged pixels,
// where it is table[bin].  diff = m*inp255 - m*matched.
// ---------------------------------------------------------------------------
__global__ void mse_kernel(const float* __restrict__ inp, const float* __restrict__ ref,
                           const float* __restrict__ msrc,
                           const unsigned int* __restrict__ flag_words,
                           const float* __restrict__ tables,
                           float* __restrict__ partials) {
    __shared__ float s_tab[3 * BINS];
    __shared__ float s_red[256];
    for (int i = threadIdx.x; i < 3 * BINS; i += blockDim.x) s_tab[i] = tables[i];
    __syncthreads();

    float acc = 0.0f;
    for (int p = blockIdx.x * blockDim.x + threadIdx.x; p < HH;
         p += gridDim.x * blockDim.x) {
        float m = msrc[p];
        bool touched = (flag_words[p >> 5] >> (p & 31)) & 1u;
#pragma unroll
        for (int c = 0; c < 3; ++c) {
            float iv = img255(inp[c * HH + p]);
            float rm = img255(ref[c * HH + p]) * m;     // ref_masked value
            float match;
            if (touched) {
                int b = (int)rm;                         // rm in [0,255]
                b = b > 255 ? 255 : (b < 0 ? 0 : b);
                match = s_tab[c * BINS + b];
            } else {
                match = rm;
            }
            float d = iv * m - match * m;
            acc += d * d;
        }
    }
    s_red[threadIdx.x] = acc;
    __syncthreads();
    for (int s = 128; s > 0; s >>= 1) {
        if (threadIdx.x < (unsigned)s) s_red[threadIdx.x] += s_red[threadIdx.x + s];
        __syncthreads();
    }
    if (threadIdx.x == 0) partials[blockIdx.x] = s_red[0];
}

// ---------------------------------------------------------------------------
// Kernel 4: deterministic final reduction (double accumulate) + mean
// ---------------------------------------------------------------------------
__global__ void finalize_kernel(const float* __restrict__ partials, int nparts,
                                float* __restrict__ out) {
    __shared__ double s[256];
    double acc = 0.0;
    for (int i = threadIdx.x; i < nparts; i += 256) acc += (double)partials[i];
    s[threadIdx.x] = acc;
    __syncthreads();
    for (int st = 128; st > 0; st >>= 1) {
        if (threadIdx.x < (unsigned)st) s[threadIdx.x] += s[threadIdx.x + st];
        __syncthreads();
    }
    if (threadIdx.x == 0)
        out[0] = (float)(s[0] / (3.0 * (double)HH));
}

extern "C" void kernel_launch(void* const* d_in, const int* in_sizes, int n_in,
                              void* d_out, int out_size, void* d_ws, size_t ws_size,
                              hipStream_t stream) {
    const float* inp  = (const float*)d_in[0];
    const float* tgt  = (const float*)d_in[1];
    const float* msrc = (const float*)d_in[2];
    const float* mtar = (const float*)d_in[3];
    const int*   i0   = (const int*)d_in[4];
    const int*   i1   = (const int*)d_in[5];
    const int*   i2   = (const int*)d_in[6];
    const int*   i3   = (const int*)d_in[7];
    const float* ref  = (const float*)d_in[8];
    const int n = in_sizes[4];

    char* ws = (char*)d_ws;
    unsigned int* g_hist     = (unsigned int*)(ws + HIST_OFF);
    float*        g_tables   = (float*)(ws + TAB_OFF);
    float*        partials   = (float*)(ws + PART_OFF);
    unsigned int* flag_words = (unsigned int*)(ws + FLAG_OFF);

    hipMemsetAsync(d_ws, 0, WS_TOTAL, stream);   // zero hists + flags each call

    hist_flags_kernel<<<2048, 256, 0, stream>>>(i0, i1, i2, i3, ref, tgt,
                                                msrc, mtar, g_hist, flag_words, n);
    cdf_table_kernel<<<1, 256, 0, stream>>>(g_hist, g_tables);
    mse_kernel<<<NPART, 256, 0, stream>>>(inp, ref, msrc, flag_words,
                                          g_tables, partials);
    finalize_kernel<<<1, 256, 0, stream>>>(partials, NPART, (float*)d_out);
}